// F_Loss_65446711656630
// MI455X (gfx1250) — compile-verified
//
#include <hip/hip_runtime.h>
#include <math.h>

#define N_ROWS 65536
#define DDIM   512
#define NCLS   16
#define NPAIRS 120          // C*(C-1)/2
#define WAVES_PER_TILE 64
#define ROWS_PER_WAVE (N_ROWS / WAVES_PER_TILE)   // 1024
#define PHASE_ROWS 32
#define NPHASES (ROWS_PER_WAVE / PHASE_ROWS)      // 32 phases per wave
#define KSTEPS_PER_PHASE (PHASE_ROWS / 4)         // 8 WMMA k-steps per phase
// dynamic LDS: [0, 32KB) stage buffers (wave*4KB, 2 x 2KB double buffer)
//              [32KB, 48KB) cross-wave reduction scratch
#define LDS_BYTES 49152

typedef __attribute__((ext_vector_type(2))) float        v2f;
typedef __attribute__((ext_vector_type(8))) float        v8f;
typedef __attribute__((ext_vector_type(4))) unsigned int v4u;
typedef __attribute__((ext_vector_type(4))) int          v4i;
typedef __attribute__((ext_vector_type(8))) int          v8i;

// ---------------------------------------------------------------------------
// Tensor Data Mover: DMA one 32-row x 16-dim f32 tile of `hidden` into LDS.
// D# per CDNA5 ISA:
//   group0: count=1 | lds_addr | global_addr[56:0] | type=2
//   group1: data_size=4B, tensor_dim0=512, tensor_dim1=65536,
//           tile_dim0=16, tile_dim1=32, tensor_dim0_stride=512
// ---------------------------------------------------------------------------
__device__ inline void tdm_load_tile(unsigned ldsByteOff, unsigned long long gaddr) {
    v4u g0;
    g0.x = 1u;                                            // count=1 (valid), user mode
    g0.y = ldsByteOff;                                    // lds_addr (bytes)
    g0.z = (unsigned)(gaddr & 0xffffffffull);             // global_addr[31:0]
    g0.w = (unsigned)((gaddr >> 32) & 0x01ffffffull)      // global_addr[56:32]
         | (2u << 30);                                    // type = 2 ("image")
    v8i g1;
    g1[0] = (2 << 16);                    // workgroup_mask=0, data_size=2 (4 bytes)
    g1[1] = (DDIM & 0xffff) << 16;        // tensor_dim0[15:0] = 512
    g1[2] = 0;                            // tensor_dim0 hi, tensor_dim1 lo (65536&0xffff=0)
    g1[3] = (N_ROWS >> 16)                // tensor_dim1[31:16] = 1
          | (16 << 16);                   // tile_dim0 = 16
    g1[4] = PHASE_ROWS;                   // tile_dim1 = 32, tile_dim2 = 0
    g1[5] = DDIM;                         // tensor_dim0_stride = 512
    g1[6] = 0;
    g1[7] = 0;
    v4i gz = {0, 0, 0, 0};
#if __has_include(<hip/amd_detail/amd_gfx1250_TDM.h>)
    v8i gz8 = {0, 0, 0, 0, 0, 0, 0, 0};
    __builtin_amdgcn_tensor_load_to_lds(g0, g1, gz, gz, gz8, 0);   // clang-23 / therock
#else
    __builtin_amdgcn_tensor_load_to_lds(g0, g1, gz, gz, 0);        // ROCm 7.2 / clang-22
#endif
}

// ---------------------------------------------------------------------------
// Kernel 0: zero workspace accumulators (ws is poisoned; atomics need zeros).
// ---------------------------------------------------------------------------
__global__ void zero_ws_kernel(float* __restrict__ ws, int n) {
    int idx = blockIdx.x * blockDim.x + threadIdx.x;
    if (idx < n) ws[idx] = 0.0f;
}

// ---------------------------------------------------------------------------
// Kernel 1: class histogram (counts as float).
// ---------------------------------------------------------------------------
__global__ void count_kernel(const int* __restrict__ bids, float* __restrict__ counts) {
    __shared__ float lc[NCLS];
    if (threadIdx.x < NCLS) lc[threadIdx.x] = 0.0f;
    __syncthreads();
    int idx = blockIdx.x * blockDim.x + threadIdx.x;
    if (idx < N_ROWS) atomicAdd(&lc[bids[idx] & (NCLS - 1)], 1.0f);
    __syncthreads();
    if (threadIdx.x < NCLS) atomicAdd(&counts[threadIdx.x], lc[threadIdx.x]);
}

// ---------------------------------------------------------------------------
// Kernel 2: segment sums + sums-of-squares.
//   D = OneHot(batch_ids)^T (16x4) x hidden-tile (4x16) + D   via
//   V_WMMA_F32_16X16X4_F32, with hidden tiles staged LDS-side by the TDM
//   (double-buffered per wave, s_wait_tensorcnt pipelining).
// A (16x4 f32): lanes 0-15 -> M=lane, K=0/1; lanes 16-31 -> M=lane-16, K=2/3.
// B (4x16 f32): lanes 0-15 -> N=lane, K=0/1; lanes 16-31 -> N=lane-16, K=2/3.
// C/D 16x16: vgpr v, lane L -> (M = v + 8*(L>=16), N = L%16).
// ---------------------------------------------------------------------------
__global__ __launch_bounds__(256) void seg_stats_wmma(
    const float* __restrict__ hidden, const int* __restrict__ bids,
    float* __restrict__ sums, float* __restrict__ sumsq) {
    extern __shared__ float lds[];

    const int tile  = blockIdx.x & 31;   // 32 dim-tiles of 16 dims
    const int cg    = blockIdx.x >> 5;   // chunk group 0..7
    const int wave  = threadIdx.x >> 5;  // 0..7
    const int lane  = threadIdx.x & 31;
    const int chunk = cg * 8 + wave;     // 0..63
    const int rowStart = chunk * ROWS_PER_WAVE;
    const int dimBase  = tile * 16;
    const int half  = lane >> 4;         // B rows: 0,1 (lanes 0-15) / 2,3 (16-31)
    const int laneM = lane & 15;         // A class row / B dim column

    const unsigned myStage = (unsigned)wave * 4096u;   // 2 x 2KB per wave
    const unsigned long long hbase =
        (unsigned long long)(size_t)hidden +
        ((unsigned long long)rowStart * DDIM + (unsigned)dimBase) * 4ull;

    // prologue: stage phase 0
    tdm_load_tile(myStage, hbase);

    v8f accS = {};
    v8f accQ = {};

    for (int p = 0; p < NPHASES; ++p) {
        if (p + 1 < NPHASES) {
            // stage next phase into the other buffer, then wait for current
            tdm_load_tile(myStage + (unsigned)((p + 1) & 1) * 2048u,
                          hbase + (unsigned long long)(p + 1) *
                                  (PHASE_ROWS * DDIM * 4ull));
            __builtin_amdgcn_s_wait_tensorcnt(1);   // in-order: phase p done
        } else {
            __builtin_amdgcn_s_wait_tensorcnt(0);
        }

        const float* buf = lds + (myStage >> 2) + (unsigned)(p & 1) * 512u;
        const int rowG = rowStart + p * PHASE_ROWS;
        __builtin_prefetch(bids + rowG + PHASE_ROWS, 0, 1);   // ids stream-ahead

        for (int s = 0; s < KSTEPS_PER_PHASE; ++s) {
            const int lr = s * 4 + half * 2;         // local row for this lane
            float b0 = buf[lr * 16 + laneM];          // conflict-free: halves of
            float b1 = buf[lr * 16 + 16 + laneM];     // wave hit disjoint banks
            const int gr = rowG + lr;
            const int iA = bids[gr];
            const int iB = bids[gr + 1];

            v2f A = { (iA == laneM) ? 1.0f : 0.0f, (iB == laneM) ? 1.0f : 0.0f };
            v2f B = { b0, b1 };
            v2f Q = { b0 * b0, b1 * b1 };

            accS = __builtin_amdgcn_wmma_f32_16x16x4_f32(false, A, false, B,
                                                         (short)0, accS, false, false);
            accQ = __builtin_amdgcn_wmma_f32_16x16x4_f32(false, A, false, Q,
                                                         (short)0, accQ, false, false);
        }
    }

    // cross-wave reduction in LDS (offset 32KB), then one atomic set per block
    float* rS = lds + 8192;          // [8][256]
    float* rQ = lds + 8192 + 2048;   // [8][256]
#pragma unroll
    for (int v = 0; v < 8; ++v) {
        rS[wave * 256 + lane * 8 + v] = accS[v];
        rQ[wave * 256 + lane * 8 + v] = accQ[v];
    }
    __syncthreads();

    const int t = threadIdx.x;       // one thread per (lane, vgpr) tile entry
    float s8 = 0.0f, q8 = 0.0f;
#pragma unroll
    for (int w = 0; w < 8; ++w) { s8 += rS[w * 256 + t]; q8 += rQ[w * 256 + t]; }
    const int eLane = t >> 3, eV = t & 7;
    const int M   = eV + ((eLane >= 16) ? 8 : 0);
    const int col = eLane & 15;
    atomicAdd(&sums [M * DDIM + dimBase + col], s8);
    atomicAdd(&sumsq[M * DDIM + dimBase + col], q8);
}

// ---------------------------------------------------------------------------
// Regularized incomplete beta I_x(a,b): Lentz continued fraction.
// ---------------------------------------------------------------------------
__device__ float betacf_dev(float a, float b, float x) {
    const float FPMIN = 1e-30f;
    float qab = a + b, qap = a + 1.0f, qam = a - 1.0f;
    float c = 1.0f;
    float d = 1.0f - qab * x / qap;
    if (fabsf(d) < FPMIN) d = FPMIN;
    d = 1.0f / d;
    float h = d;
    for (int m = 1; m <= 100; ++m) {
        float m2 = 2.0f * m;
        float aa = m * (b - m) * x / ((qam + m2) * (a + m2));
        d = 1.0f + aa * d; if (fabsf(d) < FPMIN) d = FPMIN;
        c = 1.0f + aa / c; if (fabsf(c) < FPMIN) c = FPMIN;
        d = 1.0f / d;
        h *= d * c;
        aa = -(a + m) * (qab + m) * x / ((a + m2) * (qap + m2));
        d = 1.0f + aa * d; if (fabsf(d) < FPMIN) d = FPMIN;
        c = 1.0f + aa / c; if (fabsf(c) < FPMIN) c = FPMIN;
        d = 1.0f / d;
        float del = d * c;
        h *= del;
        if (fabsf(del - 1.0f) < 3e-7f) break;
    }
    return h;
}

__device__ float betainc_dev(float a, float b, float x) {
    if (x <= 0.0f) return 0.0f;
    if (x >= 1.0f) return 1.0f;
    float lbeta = lgammaf(a + b) - lgammaf(a) - lgammaf(b);
    float front = __expf(lbeta + a * __logf(x) + b * log1pf(-x));
    if (x < (a + 1.0f) / (a + b + 2.0f))
        return front * betacf_dev(a, b, x) / a;
    else
        return 1.0f - front * betacf_dev(b, a, 1.0f - x) / b;
}

// ---------------------------------------------------------------------------
// Kernel 3: one block per class-pair: betainc over D dims, iterative top-K
// argmax in LDS, accumulate sum(log(top_k)).
// ---------------------------------------------------------------------------
__global__ __launch_bounds__(256) void pair_kernel(
    const float* __restrict__ sums, const float* __restrict__ sumsq,
    const float* __restrict__ counts, float* __restrict__ pairOut,
    const int* __restrict__ dptr) {
    __shared__ float vals[DDIM];
    __shared__ float rv[256];
    __shared__ int   ri[256];

    const int p = blockIdx.x, tid = threadIdx.x;
    int i = 0, rem = p;
    while (rem >= NCLS - 1 - i) { rem -= NCLS - 1 - i; ++i; }
    const int j = i + 1 + rem;

    const float ni = counts[i], nj = counts[j];
    const float pc = ni + nj;
    float d2 = pc - 2.0f;
    if (d2 == 0.0f) d2 = 1e-5f;
    const float a = 0.5f, b = 0.5f * d2;

    for (int k = tid; k < DDIM; k += 256) {
        float mi = sums[i * DDIM + k] / ni;
        float mj = sums[j * DDIM + k] / nj;
        float wi = sumsq[i * DDIM + k] - ni * mi * mi;
        float wj = sumsq[j * DDIM + k] - nj * mj * mj;
        float hd = 0.5f * (mi - mj);
        float bet = hd * hd * pc;
        float wit = wi + wj;
        float x = bet / (bet + wit);
        x = fminf(fmaxf(x, 1e-37f), 1.0f - 1e-5f);
        vals[k] = betainc_dev(a, b, x);
    }
    __syncthreads();

    int K = *dptr;
    if (K > DDIM) K = DDIM;
    if (K < 0) K = 0;

    float acc = 0.0f;
    for (int t = 0; t < K; ++t) {
        float v0 = vals[tid], v1 = vals[tid + 256];
        float mv = v0; int midx = tid;
        if (v1 > v0) { mv = v1; midx = tid + 256; }
        rv[tid] = mv; ri[tid] = midx;
        __syncthreads();
        for (int off = 128; off > 0; off >>= 1) {
            if (tid < off && rv[tid + off] > rv[tid]) {
                rv[tid] = rv[tid + off]; ri[tid] = ri[tid + off];
            }
            __syncthreads();
        }
        if (tid == 0) { acc += __logf(rv[0]); vals[ri[0]] = -INFINITY; }
        __syncthreads();
    }
    if (tid == 0) pairOut[p] = acc;
}

// ---------------------------------------------------------------------------
// Kernel 4: deterministic final reduction of 120 pair values -> -total.
// ---------------------------------------------------------------------------
__global__ void final_reduce(const float* __restrict__ pairOut, float* __restrict__ out) {
    __shared__ float s[128];
    const int t = threadIdx.x;
    s[t] = (t < NPAIRS) ? pairOut[t] : 0.0f;
    __syncthreads();
    for (int off = 64; off > 0; off >>= 1) {
        if (t < off) s[t] += s[t + off];
        __syncthreads();
    }
    if (t == 0) out[0] = -s[0];
}

// ---------------------------------------------------------------------------
// Workspace layout (floats):
//   [0,8192)        sums   [16][512]
//   [8192,16384)    sumsq  [16][512]
//   [16384,16400)   counts [16]
//   [16400,16520)   pairOut[120]
// ---------------------------------------------------------------------------
extern "C" void kernel_launch(void* const* d_in, const int* in_sizes, int n_in,
                              void* d_out, int out_size, void* d_ws, size_t ws_size,
                              hipStream_t stream) {
    (void)in_sizes; (void)n_in; (void)out_size; (void)ws_size;
    const float* hidden = (const float*)d_in[0];
    const int*   bids   = (const int*)d_in[1];
    const int*   dptr   = (const int*)d_in[2];
    float* out = (float*)d_out;
    float* ws  = (float*)d_ws;

    float* sums    = ws;
    float* sumsq   = ws + 8192;
    float* counts  = ws + 16384;
    float* pairOut = ws + 16400;
    const int ws_floats = 16520;

    zero_ws_kernel<<<(ws_floats + 255) / 256, 256, 0, stream>>>(ws, ws_floats);
    count_kernel<<<N_ROWS / 256, 256, 0, stream>>>(bids, counts);
    seg_stats_wmma<<<256, 256, LDS_BYTES, stream>>>(hidden, bids, sums, sumsq);
    pair_kernel<<<NPAIRS, 256, 0, stream>>>(sums, sumsq, counts, pairOut, dptr);
    final_reduce<<<1, 128, 0, stream>>>(pairOut, out);
}